// Attention_79216376807935
// MI455X (gfx1250) — compile-verified
//
#include <hip/hip_runtime.h>
#include <hip/hip_bf16.h>

#define DD 256
#define KK 32
#define NSTEPS 2
#define BLK 256
#define RSTRIDE (DD + 4)   // 260 floats: pad => conflict-free WMMA fragment loads
#define NEG_SLOPE 0.2f

typedef __attribute__((ext_vector_type(2))) float v2f;
typedef __attribute__((ext_vector_type(8))) float v8f;
typedef int v4i __attribute__((vector_size(16)));

#if __has_builtin(__builtin_amdgcn_wmma_f32_16x16x4_f32)
#define USE_WMMA 1
#endif
#if __has_builtin(__builtin_amdgcn_global_load_async_to_lds_b128) && \
    __has_builtin(__builtin_amdgcn_s_wait_asynccnt)
#define USE_ASYNC 1
#endif

__global__ __launch_bounds__(BLK) void gat_agg_kernel(
    const int* __restrict__ node,
    const int* __restrict__ neighbors,
    const float* __restrict__ emb,
    const float* __restrict__ fc_w,
    const float* __restrict__ fc_b,
    float* __restrict__ out,
    int n_nodes)
{
  __shared__ __align__(16) float nb[NSTEPS][KK * RSTRIDE];  // ~65 KB, double buffer
  __shared__ float wnb[DD];
  __shared__ float wself[DD];
  __shared__ int   idx[NSTEPS][KK];
  __shared__ float scorepart[8][KK];
  __shared__ float attn[KK];
  __shared__ float selfsc;
  __shared__ float wred[8];

  const int tid = threadIdx.x;
  const int b   = blockIdx.x;
  const int n   = node[b];

  // Stage weights + neighbor indices for both hop-steps.
  wnb[tid]   = fc_w[tid];
  wself[tid] = fc_w[DD + tid];
  if (tid < NSTEPS * KK) {
    const int s = tid >> 5, k = tid & 31;
    idx[s][k] = neighbors[(size_t)s * n_nodes * KK + (size_t)n * KK + k];
  }
  const float e_self = emb[(size_t)n * DD + tid];
  __syncthreads();

  // self score = dot(node_e, w_self): wave reduce then cross-wave reduce
  {
    float p = e_self * wself[tid];
    for (int off = 16; off; off >>= 1) p += __shfl_xor(p, off, 32);
    if ((tid & 31) == 0) wred[tid >> 5] = p;
  }
  __syncthreads();
  if (tid == 0) {
    float t = 0.f;
    #pragma unroll
    for (int w = 0; w < 8; ++w) t += wred[w];
    selfsc = t + fc_b[0];
  }

  // ---- Gather 2 x (32 rows x 1KB) into LDS. Thread owns a fixed 16B column
  // across 8 rows; a wave covers a coalesced 512B half-row per instruction.
  const int colf4 = tid & 63;   // 16B chunk within a row
  const int r0    = tid >> 6;   // base row (0..3)
#ifdef USE_ASYNC
  #pragma unroll
  for (int s = 0; s < NSTEPS; ++s) {
    #pragma unroll
    for (int j = 0; j < 8; ++j) {
      const int r = r0 + 4 * j;
      const float* g = emb + (size_t)idx[s][r] * DD + colf4 * 4;
      float* l = &nb[s][r * RSTRIDE + colf4 * 4];
      __builtin_amdgcn_global_load_async_to_lds_b128(
          (__attribute__((address_space(1))) v4i*)
              ((__attribute__((address_space(1))) void*)g),
          (__attribute__((address_space(3))) v4i*)
              ((__attribute__((address_space(3))) void*)l),
          0, 0);
    }
  }
#else
  #pragma unroll
  for (int s = 0; s < NSTEPS; ++s)
    #pragma unroll
    for (int j = 0; j < 8; ++j) {
      const int r = r0 + 4 * j;
      const float4 v = *(const float4*)(emb + (size_t)idx[s][r] * DD + colf4 * 4);
      *(float4*)&nb[s][r * RSTRIDE + colf4 * 4] = v;
    }
#endif

  const int lane = tid & 31, wv = tid >> 5;
  const int m = lane & 15;
  const bool hi = lane >= 16;

#ifdef USE_WMMA
  // B fragments (w_nb in column N=0) are invariant across the s-loop: build
  // once. Unconditional ds_load + multiply-mask (w*0 != 0 for NaN under strict
  // FP, so clang cannot re-predicate the load) => no saveexec in the hot loop.
  const float bmsk = (m == 0) ? 1.f : 0.f;
  v2f bfr[8];
  #pragma unroll
  for (int st = 0; st < 8; ++st) {
    const int kk = wv * 32 + st * 4 + (hi ? 2 : 0);
    const float2 w = *(const float2*)&wnb[kk];
    bfr[st][0] = w.x * bmsk;
    bfr[st][1] = w.y * bmsk;
  }
#endif

  float acc = 0.f;
  for (int s = 0; s < NSTEPS; ++s) {
#ifdef USE_ASYNC
    if (s == 0) __builtin_amdgcn_s_wait_asynccnt(8);  // buffer 0 done, buffer 1 in flight
    else        __builtin_amdgcn_s_wait_asynccnt(0);
#endif
    __syncthreads();
    const float* rowbuf = nb[s];

    // ---- scores[k] = dot(nb_row_k, w_nb): WMMA, each wave owns a 32-wide K-slice
#ifdef USE_WMMA
    {
      v8f c0 = {}; v8f c1 = {};
      #pragma unroll
      for (int st = 0; st < 8; ++st) {
        const int kk = wv * 32 + st * 4 + (hi ? 2 : 0);
        // A fragments: 16x4 f32 tile; lanes 0-15 K={0,1}, lanes 16-31 K={2,3}
        const float2 fa0 = *(const float2*)&rowbuf[m * RSTRIDE + kk];
        const float2 fa1 = *(const float2*)&rowbuf[(m + 16) * RSTRIDE + kk];
        v2f a0; a0[0] = fa0.x; a0[1] = fa0.y;
        v2f a1; a1[0] = fa1.x; a1[1] = fa1.y;
        c0 = __builtin_amdgcn_wmma_f32_16x16x4_f32(false, a0, false, bfr[st],
                                                   (short)0, c0, false, false);
        c1 = __builtin_amdgcn_wmma_f32_16x16x4_f32(false, a1, false, bfr[st],
                                                   (short)0, c1, false, false);
      }
      // column N=0 of C lives in lanes 0 (M=0..7) and 16 (M=8..15)
      if (m == 0) {
        const int rb = hi ? 8 : 0;
        #pragma unroll
        for (int r = 0; r < 8; ++r) {
          scorepart[wv][rb + r]      = c0[r];
          scorepart[wv][16 + rb + r] = c1[r];
        }
      }
    }
#else
    {
      #pragma unroll
      for (int rr = 0; rr < 4; ++rr) {
        const int r = wv * 4 + rr;
        float p = 0.f;
        for (int dd = lane; dd < DD; dd += 32) p += rowbuf[r * RSTRIDE + dd] * wnb[dd];
        for (int off = 16; off; off >>= 1) p += __shfl_xor(p, off, 32);
        if (lane == 0) scorepart[0][r] = p;
      }
    }
#endif
    __syncthreads();

    // ---- leaky-relu + softmax over K, wave 0, lane k owns score k
    if (tid < KK) {
      float sc;
#ifdef USE_WMMA
      sc = 0.f;
      #pragma unroll
      for (int w = 0; w < 8; ++w) sc += scorepart[w][tid];
#else
      sc = scorepart[0][tid];
#endif
      sc += selfsc;
      sc = (sc >= 0.f) ? sc : NEG_SLOPE * sc;
      float mx = sc;
      for (int off = 16; off; off >>= 1) mx = fmaxf(mx, __shfl_xor(mx, off, 32));
      const float e = __expf(sc - mx);
      float sm = e;
      for (int off = 16; off; off >>= 1) sm += __shfl_xor(sm, off, 32);
      attn[tid] = e / sm;
    }
    __syncthreads();

    // ---- aggregate: lane = feature d; broadcast attn, conflict-free row reads
    #pragma unroll
    for (int k = 0; k < KK; ++k)
      acc = fmaf(attn[k], rowbuf[k * RSTRIDE + tid], acc);
  }

  out[(size_t)b * DD + tid] = acc + (float)(NSTEPS * KK) * e_self;
}

extern "C" void kernel_launch(void* const* d_in, const int* in_sizes, int n_in,
                              void* d_out, int out_size, void* d_ws, size_t ws_size,
                              hipStream_t stream) {
  const int*   node      = (const int*)d_in[0];
  const int*   neighbors = (const int*)d_in[1];
  const float* emb       = (const float*)d_in[2];
  const float* fc_w      = (const float*)d_in[3];
  const float* fc_b      = (const float*)d_in[4];
  float*       out       = (float*)d_out;
  const int B       = in_sizes[0];
  const int n_nodes = in_sizes[2] / DD;
  gat_agg_kernel<<<B, BLK, 0, stream>>>(node, neighbors, emb, fc_w, fc_b, out, n_nodes);
}